// LinearConv2D_78941498900730
// MI455X (gfx1250) — compile-verified
//
#include <hip/hip_runtime.h>

// ---------------------------------------------------------------------------
// LinearConv2D for MI455X (gfx1250): fused unfold + grouped einsum + freq
// box-sum, expressed as 512 small GEMMs (M=976, N=8->16, K=256) on the fp32
// WMMA pipe (v_wmma_f32_16x16x4_f32). x is staged into LDS with the CDNA5
// async path (global_load_async_to_lds_b128, ASYNCcnt), B fragments are
// shared across two time-tiles per wave (dual accumulators).
// ---------------------------------------------------------------------------

typedef float v2f __attribute__((ext_vector_type(2)));
typedef float v8f __attribute__((ext_vector_type(8)));

// Problem constants (match reference)
constexpr int Bn = 16;    // batch
constexpr int Cc = 64;    // input channels
constexpr int Fn = 128;   // freq bins
constexpr int Tn = 256;   // time samples
constexpr int Gg = 8;     // groups
constexpr int Nn = 8;     // filters / group
constexpr int Oo = 64;    // out channels
constexpr int Tt = 61;    // output time steps
constexpr int FP = 64;    // output freq pairs

// Tiling
constexpr int CH      = 16;   // (fo,d) combos -> K/16
constexpr int TWIN    = 48;   // time window per chunk: 8 tiles*4 + 12 halo + 4 pad
constexpr int XPITCH  = 52;   // LDS row pitch for x: 16B-aligned, bank-spread
constexpr int WPITCH  = 260;  // LDS row pitch for weights: /4, bank-spread
constexpr int TILES_PER_BLK = 8;

__global__ __launch_bounds__(128)
void lconv2d_wmma_f32(const float* __restrict__ xg,
                      const float* __restrict__ wt,
                      float* __restrict__ out)
{
    // xs[(fo*8+d)*16 + b][tt] , tt in [0,48), pitch 52
    __shared__ __align__(16) float xs[CH * 16 * XPITCH];
    // ws[n][k] for n in [0,8), plus row 8 == all zeros (N padding cols 8..15)
    __shared__ __align__(16) float ws[9 * WPITCH];

    const int tid   = threadIdx.x;
    const int bx    = blockIdx.x;
    const int fp    = bx & 63;          // freq pair
    const int g     = (bx >> 6) & 7;    // group
    const int chunk = bx >> 9;          // 0..7 (8 output-time tiles each)

    // ---- stage weights: ws[n][(fo*8+d)*16 + w] = wt[(8g+n), d, 2fp+fo, w]
    {
        const int n  = tid >> 4;        // 0..7
        const int d  = (tid >> 1) & 7;  // 0..7
        const int fo = tid & 1;         // 0..1
        const float4* src =
            (const float4*)(wt + ((((g * 8 + n) * 8 + d) * Fn) + 2 * fp + fo) * 16);
        float4* dst = (float4*)(ws + n * WPITCH + (fo * 8 + d) * 16);
        dst[0] = src[0]; dst[1] = src[1]; dst[2] = src[2]; dst[3] = src[3];
        // zero row 8 (read by padded B columns 8..15)
        float4* zp = (float4*)(ws + 8 * WPITCH);
        if (tid < 64) zp[tid] = float4{0.f, 0.f, 0.f, 0.f};
    }

    // ---- stage x slice asynchronously (ASYNCcnt path, LDS written directly):
    //      xs[(fo*8+d)*16 + b][tt] = x[b, 8g+d, 2fp+fo, chunk*32+tt]
    {
        for (int q = tid; q < CH * 16 * (TWIN / 4); q += 128) {
            const int tt4 = q % (TWIN / 4);          // float4 index in window
            const int rem = q / (TWIN / 4);
            const int b   = rem & 15;
            const int c16 = rem >> 4;                // fo*8+d
            const int d   = c16 & 7;
            const int fo  = c16 >> 3;
            int ts = chunk * 32 + tt4 * 4;
            if (ts > Tn - 4) ts = Tn - 4;            // clamp (tail never read)
            const float* gp =
                xg + (((b * Cc + g * 8 + d) * Fn) + 2 * fp + fo) * Tn + ts;
            unsigned int loff =
                (unsigned int)(uintptr_t)(xs + (c16 * 16 + b) * XPITCH + tt4 * 4);
            unsigned long long ga = (unsigned long long)(uintptr_t)gp;
            // 16B global src, 16B LDS dst; tracked by ASYNCcnt
            asm volatile("global_load_async_to_lds_b128 %0, %1, off"
                         :: "v"(loff), "v"(ga)
                         : "memory");
        }
#if defined(__has_builtin) && __has_builtin(__builtin_amdgcn_s_wait_asynccnt)
        __builtin_amdgcn_s_wait_asynccnt(0);
#else
        asm volatile("s_wait_asynccnt 0x0" ::: "memory");
#endif
    }
    __syncthreads();

    const int wave = tid >> 5;
    const int lane = tid & 31;
    const int col  = lane & 15;          // N column / A row (batch index base)
    const int half = lane >> 4;          // 0: K={k0,k0+1}, 1: K={k0+2,k0+3}
    const int brow = (col < Nn) ? col : 8;  // padded columns read the zero row

    // Each wave owns two adjacent time-tiles; B fragments shared between them.
    const int tb0 = 2 * wave;            // 0,2,4,6
    const int t0  = chunk * TILES_PER_BLK + tb0;
    const int t1  = t0 + 1;

    if (t0 < Tt) {                       // wave-uniform
        v8f acc0 = {0.f,0.f,0.f,0.f,0.f,0.f,0.f,0.f};
        v8f acc1 = {0.f,0.f,0.f,0.f,0.f,0.f,0.f,0.f};
        const float* ap0 = xs + col * XPITCH + tb0 * 4 + 2 * half;
        const float* bp0 = ws + brow * WPITCH + 2 * half;

        #pragma unroll
        for (int c16 = 0; c16 < CH; ++c16) {
            const float* ap = ap0 + c16 * (16 * XPITCH);
            const float* bp = bp0 + c16 * 16;
            #pragma unroll
            for (int jw = 0; jw < 4; ++jw) {     // k0 = (c16*4 + jw)*4
                v2f b  = *(const v2f*)(bp + jw * 4);       // B[k0+2h.., col]
                v2f a0 = *(const v2f*)(ap + jw * 4);       // tile t0
                v2f a1 = *(const v2f*)(ap + jw * 4 + 4);   // tile t1 (+1 step)
                acc0 = __builtin_amdgcn_wmma_f32_16x16x4_f32(
                    false, a0, false, b, (short)0, acc0, false, false);
                acc1 = __builtin_amdgcn_wmma_f32_16x16x4_f32(
                    false, a1, false, b, (short)0, acc1, false, false);
            }
        }

        if (col < Nn) {
            #pragma unroll
            for (int v = 0; v < 8; ++v) {
                const int b = v + 8 * half;                 // D row M = batch
                out[(((b * Oo) + g * 8 + col) * FP + fp) * Tt + t0] = acc0[v];
            }
            if (t1 < Tt) {
                #pragma unroll
                for (int v = 0; v < 8; ++v) {
                    const int b = v + 8 * half;
                    out[(((b * Oo) + g * 8 + col) * FP + fp) * Tt + t1] = acc1[v];
                }
            }
        }
    }
}

extern "C" void kernel_launch(void* const* d_in, const int* in_sizes, int n_in,
                              void* d_out, int out_size, void* d_ws, size_t ws_size,
                              hipStream_t stream) {
    const float* x  = (const float*)d_in[0];   // [16,64,128,256] f32
    const float* wt = (const float*)d_in[1];   // [64,8,128,16]  f32
    float* out = (float*)d_out;                // [16,64,64,61]  f32
    (void)in_sizes; (void)n_in; (void)d_ws; (void)ws_size; (void)out_size;

    // 8 groups x 64 freq-pairs x 8 time-chunks
    dim3 grid(Gg * FP * 8);
    dim3 block(128);
    hipLaunchKernelGGL(lconv2d_wmma_f32, grid, block, 0, stream, x, wt, out);
}